// AgnosticResidualInteractionBlock_44676249813161
// MI455X (gfx1250) — compile-verified
//
#include <hip/hip_runtime.h>

// ---------------------------------------------------------------------------
// MI455X (gfx1250) implementation of the AgnosticResidualInteractionBlock.
// fp32 throughout (matches reference numerics) using V_WMMA_F32_16X16X4_F32.
// K-outer / tile-inner GEMM loops: one A-fragment feeds 4..16 WMMAs.
// Optional CDNA5 async global->LDS staging (GLOBAL_LOAD_ASYNC_TO_LDS_B128).
// ---------------------------------------------------------------------------

typedef float v2f __attribute__((ext_vector_type(2)));
typedef float v8f __attribute__((ext_vector_type(8)));
typedef int   v4i __attribute__((vector_size(16)));

#define NATTRC 10

// ---- CDNA5 async global->LDS (ASYNCcnt) path, feature-guarded ------------
#if __has_builtin(__builtin_amdgcn_global_load_async_to_lds_b128) && \
    __has_builtin(__builtin_amdgcn_s_wait_asynccnt)
#define USE_ASYNC_LDS 1
#else
#define USE_ASYNC_LDS 0
#endif

#if USE_ASYNC_LDS
__device__ __forceinline__ void async_g2l_b128(void* lds, const void* g) {
  // signature (from compiler diagnostic): param0 = generic v4i* (global src)
  __builtin_amdgcn_global_load_async_to_lds_b128(
      (v4i*)const_cast<void*>(g),
      (__attribute__((address_space(3))) v4i*)(unsigned int)(unsigned long long)(lds),
      0, 0);
}
__device__ __forceinline__ void async_wait0() {
  __builtin_amdgcn_s_wait_asynccnt(0);
}
#endif

__device__ __forceinline__ v8f zero8() {
  v8f z = {0.f,0.f,0.f,0.f,0.f,0.f,0.f,0.f};
  return z;
}

__device__ __forceinline__ v8f wmma4(v2f a, v2f b, v8f c) {
  // D = A(16x4) * B(4x16) + C(16x16), all fp32, wave32.
  return __builtin_amdgcn_wmma_f32_16x16x4_f32(false, a, false, b, (short)0, c,
                                               false, false);
}

__device__ __forceinline__ float silu(float x) {
  return x / (1.0f + __expf(-x));
}

__device__ __forceinline__ void atomAdd(float* p, float v) {
  // native global_atomic_add_f32 (accumulators are coarse-grained HBM)
  unsafeAtomicAdd(p, v);
}

// Constants
__device__ const float SK_NORM  = 0.0395284707521047f;   // 1/sqrt(64*10)
__device__ const float UP_NORM  = 0.125f;                // 1/sqrt(64)
__device__ const float L1_SCALE = 0.3535533905932738f;   // 1/sqrt(8)
__device__ const float L_SCALE  = 0.125f;                // 1/8
__device__ const float LIN_NORM = 0.0027621358640100f;   // 1/(sqrt(128)*32)
__device__ const float INV3C    = 0.5773502691896258f;

// ---------------------------------------------------------------------------
// Kernel 1: node prep.
//   sc_s (N,128), sc_v (N,64,3)  : K=640, A[n,k]=x[n,k/10]*attr[n,k%10]
//   x0u  (N,64),  x1u  (N,64,3)  : K=64 up-projections
// One wave handles 16 nodes. 2 waves / block.
// ---------------------------------------------------------------------------
__global__ __launch_bounds__(64) void k1_node_prep(
    const float* __restrict__ node_attrs, const float* __restrict__ node_feats,
    const float* __restrict__ W_up0, const float* __restrict__ W_up1,
    const float* __restrict__ W_sk_s, const float* __restrict__ W_sk_v,
    float* __restrict__ x0u, float* __restrict__ x1u,
    float* __restrict__ sc_s, float* __restrict__ sc_v, int N)
{
  __shared__ __align__(16) float shF[2][16][256];
  __shared__ float shA[2][16][NATTRC];

  const int w    = threadIdx.x >> 5;
  const int lane = threadIdx.x & 31;
  const int tile = blockIdx.x * 2 + w;
  const int nT   = (N + 15) >> 4;

  // Stage 16 nodes (clamped) into this wave's LDS slab (b128 traffic).
  {
    int t = (tile < nT) ? tile : (nT - 1);
    int base = t * 16;
    const float4* src = reinterpret_cast<const float4*>(node_feats);
    for (int idx = lane; idx < 16 * 64; idx += 32) {     // 16 rows x 64 float4
      int m = idx >> 6, c4 = idx & 63;
      int nn = base + m; if (nn >= N) nn = N - 1;
#if USE_ASYNC_LDS
      async_g2l_b128(&shF[w][m][c4 * 4], &src[nn * 64 + c4]);
#else
      reinterpret_cast<float4*>(&shF[w][m][0])[c4] = src[nn * 64 + c4];
#endif
    }
    for (int idx = lane; idx < 16 * NATTRC; idx += 32) {
      int m = idx / NATTRC, a = idx - m * NATTRC;
      int nn = base + m; if (nn >= N) nn = N - 1;
      shA[w][m][a] = node_attrs[nn * NATTRC + a];
    }
  }
#if USE_ASYNC_LDS
  async_wait0();
#endif
  __syncthreads();
  if (tile >= nT) return;

  const int n0    = tile * 16;
  const int col   = lane & 15;
  const int kh    = (lane >> 4) * 2;   // K sub-offset within a 4-block
  const int mrow  = lane & 15;         // A-fragment row
  const int rbase = (lane >> 4) * 8;   // C/D row base
  const bool full = (n0 + 16 <= N);

  // ---- sc_s : (16x640) @ (640x128), K-outer, 8 tiles ----
  {
    v8f acc[8];
#pragma unroll
    for (int t = 0; t < 8; ++t) acc[t] = zero8();
    int u0 = 0, a0 = kh;                  // k0 = 4*kb + kh
    for (int kb = 0; kb < 160; ++kb) {
      int a1 = a0 + 1, u1 = u0;
      if (a1 >= NATTRC) { a1 -= NATTRC; u1 = u0 + 1; }
      v2f a;
      a.x = shF[w][mrow][u0] * shA[w][mrow][a0];
      a.y = shF[w][mrow][u1] * shA[w][mrow][a1];
      const float* Bp = W_sk_s + (kb * 4 + kh) * 128 + col;
#pragma unroll
      for (int t = 0; t < 8; ++t) {
        v2f b; b.x = Bp[t * 16]; b.y = Bp[128 + t * 16];
        acc[t] = wmma4(a, b, acc[t]);
      }
      a0 += 4; if (a0 >= NATTRC) { a0 -= NATTRC; ++u0; }
    }
#pragma unroll
    for (int t = 0; t < 8; ++t) {
      if (full) {
#pragma unroll
        for (int r = 0; r < 8; ++r)
          sc_s[(n0 + rbase + r) * 128 + t * 16 + col] = acc[t][r] * SK_NORM;
      } else {
#pragma unroll
        for (int r = 0; r < 8; ++r) {
          int nn = n0 + rbase + r;
          if (nn < N) sc_s[nn * 128 + t * 16 + col] = acc[t][r] * SK_NORM;
        }
      }
    }
  }

  // ---- sc_v : per i, (16x640) @ (640x64), K-outer, 4 tiles ----
  for (int i = 0; i < 3; ++i) {
    v8f acc[4];
#pragma unroll
    for (int t = 0; t < 4; ++t) acc[t] = zero8();
    int u0 = 0, a0 = kh;
    for (int kb = 0; kb < 160; ++kb) {
      int a1 = a0 + 1, u1 = u0;
      if (a1 >= NATTRC) { a1 -= NATTRC; u1 = u0 + 1; }
      v2f a;
      a.x = shF[w][mrow][64 + u0 * 3 + i] * shA[w][mrow][a0];
      a.y = shF[w][mrow][64 + u1 * 3 + i] * shA[w][mrow][a1];
      const float* Bp = W_sk_v + (kb * 4 + kh) * 64 + col;
#pragma unroll
      for (int t = 0; t < 4; ++t) {
        v2f b; b.x = Bp[t * 16]; b.y = Bp[64 + t * 16];
        acc[t] = wmma4(a, b, acc[t]);
      }
      a0 += 4; if (a0 >= NATTRC) { a0 -= NATTRC; ++u0; }
    }
#pragma unroll
    for (int t = 0; t < 4; ++t) {
      if (full) {
#pragma unroll
        for (int r = 0; r < 8; ++r)
          sc_v[((n0 + rbase + r) * 64 + t * 16 + col) * 3 + i] = acc[t][r] * SK_NORM;
      } else {
#pragma unroll
        for (int r = 0; r < 8; ++r) {
          int nn = n0 + rbase + r;
          if (nn < N) sc_v[(nn * 64 + t * 16 + col) * 3 + i] = acc[t][r] * SK_NORM;
        }
      }
    }
  }

  // ---- x0u : (16x64) @ W_up0(64x64), K-outer, 4 tiles ----
  {
    v8f acc[4];
#pragma unroll
    for (int t = 0; t < 4; ++t) acc[t] = zero8();
    for (int kb = 0; kb < 16; ++kb) {
      int k0 = kb * 4 + kh;
      v2f a;
      a.x = shF[w][mrow][k0];
      a.y = shF[w][mrow][k0 + 1];
      const float* Bp = W_up0 + k0 * 64 + col;
#pragma unroll
      for (int t = 0; t < 4; ++t) {
        v2f b; b.x = Bp[t * 16]; b.y = Bp[64 + t * 16];
        acc[t] = wmma4(a, b, acc[t]);
      }
    }
#pragma unroll
    for (int t = 0; t < 4; ++t) {
      if (full) {
#pragma unroll
        for (int r = 0; r < 8; ++r)
          x0u[(n0 + rbase + r) * 64 + t * 16 + col] = acc[t][r] * UP_NORM;
      } else {
#pragma unroll
        for (int r = 0; r < 8; ++r) {
          int nn = n0 + rbase + r;
          if (nn < N) x0u[nn * 64 + t * 16 + col] = acc[t][r] * UP_NORM;
        }
      }
    }
  }

  // ---- x1u : per i, (16x64) @ W_up1(64x64), K-outer, 4 tiles ----
  for (int i = 0; i < 3; ++i) {
    v8f acc[4];
#pragma unroll
    for (int t = 0; t < 4; ++t) acc[t] = zero8();
    for (int kb = 0; kb < 16; ++kb) {
      int k0 = kb * 4 + kh;
      v2f a;
      a.x = shF[w][mrow][64 + k0 * 3 + i];
      a.y = shF[w][mrow][64 + (k0 + 1) * 3 + i];
      const float* Bp = W_up1 + k0 * 64 + col;
#pragma unroll
      for (int t = 0; t < 4; ++t) {
        v2f b; b.x = Bp[t * 16]; b.y = Bp[64 + t * 16];
        acc[t] = wmma4(a, b, acc[t]);
      }
    }
#pragma unroll
    for (int t = 0; t < 4; ++t) {
      if (full) {
#pragma unroll
        for (int r = 0; r < 8; ++r)
          x1u[((n0 + rbase + r) * 64 + t * 16 + col) * 3 + i] = acc[t][r] * UP_NORM;
      } else {
#pragma unroll
        for (int r = 0; r < 8; ++r) {
          int nn = n0 + rbase + r;
          if (nn < N) x1u[(nn * 64 + t * 16 + col) * 3 + i] = acc[t][r] * UP_NORM;
        }
      }
    }
  }
}

// ---------------------------------------------------------------------------
// Kernel 2: edge MLP (WMMA, LDS round-trip between layers) + tensor-product
// message construction + atomic scatter into receiver accumulators.
// One wave per 16 edges, 2 waves / block.
// ---------------------------------------------------------------------------
__global__ __launch_bounds__(64) void k2_edge(
    const float* __restrict__ edge_feats,
    const float* __restrict__ ear, const float* __restrict__ eai,
    const int* __restrict__ eidx,
    const float* __restrict__ M1, const float* __restrict__ M2,
    const float* __restrict__ M3, const float* __restrict__ M4,
    const float* __restrict__ x0u, const float* __restrict__ x1u,
    float* __restrict__ msr, float* __restrict__ mvr,
    float* __restrict__ msi, float* __restrict__ mvi,
    int N, int E)
{
  __shared__ __align__(16) float ef[2][16][8];
  __shared__ int   snd[2][16], rcv[2][16];
  __shared__ __align__(16) float yr[2][16][4], yi[2][16][4];
  __shared__ __align__(16) float hA[2][16][64], hB[2][16][64];
  __shared__ __align__(16) float tw[2][16][256];

  const int w    = threadIdx.x >> 5;
  const int lane = threadIdx.x & 31;
  const int tile = blockIdx.x * 2 + w;
  const int nT   = (E + 15) >> 4;
  const bool valid = (tile < nT);
  const int e0   = (valid ? tile : 0) * 16;

  // ---- stage edge meta + features (b128, async where available) ----
  {
    const float4* efs = reinterpret_cast<const float4*>(edge_feats);
    const float4* yrs = reinterpret_cast<const float4*>(ear);
    const float4* yis = reinterpret_cast<const float4*>(eai);
    for (int idx = lane; idx < 16 * 2; idx += 32) {       // 16 rows x 2 float4
      int m = idx >> 1, c4 = idx & 1;
      int e = e0 + m; if (e >= E) e = E - 1;
#if USE_ASYNC_LDS
      async_g2l_b128(&ef[w][m][c4 * 4], &efs[e * 2 + c4]);
#else
      reinterpret_cast<float4*>(&ef[w][m][0])[c4] = efs[e * 2 + c4];
#endif
    }
    if (lane < 16) {
      int e = e0 + lane; if (e >= E) e = E - 1;
#if USE_ASYNC_LDS
      async_g2l_b128(&yr[w][lane][0], &yrs[e]);
      async_g2l_b128(&yi[w][lane][0], &yis[e]);
#else
      reinterpret_cast<float4*>(&yr[w][lane][0])[0] = yrs[e];
      reinterpret_cast<float4*>(&yi[w][lane][0])[0] = yis[e];
#endif
      snd[w][lane] = eidx[e];
      rcv[w][lane] = eidx[E + e];
    }
  }
#if USE_ASYNC_LDS
  async_wait0();
#endif
  __syncthreads();

  const int col   = lane & 15;
  const int kh    = (lane >> 4) * 2;
  const int mrow  = lane & 15;
  const int rbase = (lane >> 4) * 8;

  // ---- layer 1: silu(ef @ M1 / sqrt(8))  (16x8)@(8x64) -> hA ----
  {
    v8f acc[4];
#pragma unroll
    for (int t = 0; t < 4; ++t) acc[t] = zero8();
#pragma unroll
    for (int kb = 0; kb < 2; ++kb) {
      int k0 = kb * 4 + kh;
      v2f a;
      a.x = ef[w][mrow][k0]; a.y = ef[w][mrow][k0 + 1];
      const float* Bp = M1 + k0 * 64 + col;
#pragma unroll
      for (int t = 0; t < 4; ++t) {
        v2f b; b.x = Bp[t * 16]; b.y = Bp[64 + t * 16];
        acc[t] = wmma4(a, b, acc[t]);
      }
    }
#pragma unroll
    for (int t = 0; t < 4; ++t)
#pragma unroll
      for (int r = 0; r < 8; ++r)
        hA[w][rbase + r][t * 16 + col] = silu(acc[t][r] * L1_SCALE);
  }
  __syncthreads();

  // ---- layer 2: silu(hA @ M2 / 8) -> hB ----
  {
    v8f acc[4];
#pragma unroll
    for (int t = 0; t < 4; ++t) acc[t] = zero8();
    for (int kb = 0; kb < 16; ++kb) {
      int k0 = kb * 4 + kh;
      v2f a;
      a.x = hA[w][mrow][k0]; a.y = hA[w][mrow][k0 + 1];
      const float* Bp = M2 + k0 * 64 + col;
#pragma unroll
      for (int t = 0; t < 4; ++t) {
        v2f b; b.x = Bp[t * 16]; b.y = Bp[64 + t * 16];
        acc[t] = wmma4(a, b, acc[t]);
      }
    }
#pragma unroll
    for (int t = 0; t < 4; ++t)
#pragma unroll
      for (int r = 0; r < 8; ++r)
        hB[w][rbase + r][t * 16 + col] = silu(acc[t][r] * L_SCALE);
  }
  __syncthreads();

  // ---- layer 3: silu(hB @ M3 / 8) -> hA ----
  {
    v8f acc[4];
#pragma unroll
    for (int t = 0; t < 4; ++t) acc[t] = zero8();
    for (int kb = 0; kb < 16; ++kb) {
      int k0 = kb * 4 + kh;
      v2f a;
      a.x = hB[w][mrow][k0]; a.y = hB[w][mrow][k0 + 1];
      const float* Bp = M3 + k0 * 64 + col;
#pragma unroll
      for (int t = 0; t < 4; ++t) {
        v2f b; b.x = Bp[t * 16]; b.y = Bp[64 + t * 16];
        acc[t] = wmma4(a, b, acc[t]);
      }
    }
#pragma unroll
    for (int t = 0; t < 4; ++t)
#pragma unroll
      for (int r = 0; r < 8; ++r)
        hA[w][rbase + r][t * 16 + col] = silu(acc[t][r] * L_SCALE);
  }
  __syncthreads();

  // ---- layer 4: tw = hA @ M4 / 8   (16x64)@(64x256), 16 tiles ----
  {
    v8f acc[16];
#pragma unroll
    for (int t = 0; t < 16; ++t) acc[t] = zero8();
    for (int kb = 0; kb < 16; ++kb) {
      int k0 = kb * 4 + kh;
      v2f a;
      a.x = hA[w][mrow][k0]; a.y = hA[w][mrow][k0 + 1];
      const float* Bp = M4 + k0 * 256 + col;
#pragma unroll
      for (int t = 0; t < 16; ++t) {
        v2f b; b.x = Bp[t * 16]; b.y = Bp[256 + t * 16];
        acc[t] = wmma4(a, b, acc[t]);
      }
    }
#pragma unroll
    for (int t = 0; t < 16; ++t)
#pragma unroll
      for (int r = 0; r < 8; ++r)
        tw[w][rbase + r][t * 16 + col] = acc[t][r] * L_SCALE;
  }
  __syncthreads();

  if (!valid) return;   // no barriers below this point

  // ---- tensor-product messages + scatter (segment_sum via L2 fp32 atomics)
  for (int nb = 0; nb < 4; ++nb) {
    int u = nb * 16 + col;
#pragma unroll
    for (int r = 0; r < 8; ++r) {
      int m = rbase + r;
      if (e0 + m >= E) continue;
      int sn = snd[w][m], rc = rcv[w][m];
      float x0v = x0u[sn * 64 + u];
      float wAv = tw[w][m][u];
      float wBv = tw[w][m][64 + u];
      float y0r  = yr[w][m][0];
      float y0im = yi[w][m][0];
      atomAdd(&msr[rc * 128 + u], x0v * y0r * wAv);
      atomAdd(&msi[rc * 128 + u], x0v * y0im * wAv);
      float bc = x0v * wBv;
#pragma unroll
      for (int i = 0; i < 3; ++i) {
        atomAdd(&mvr[(rc * 128 + u) * 3 + i], bc * yr[w][m][1 + i]);
        atomAdd(&mvi[(rc * 128 + u) * 3 + i], bc * yi[w][m][1 + i]);
      }
    }
  }
  for (int nb = 0; nb < 4; ++nb) {
    int u = nb * 16 + col;
#pragma unroll
    for (int r = 0; r < 8; ++r) {
      int m = rbase + r;
      if (e0 + m >= E) continue;
      int sn = snd[w][m], rc = rcv[w][m];
      float wCv = tw[w][m][128 + u];
      float wDv = tw[w][m][192 + u];
      float x1v0 = x1u[(sn * 64 + u) * 3 + 0];
      float x1v1 = x1u[(sn * 64 + u) * 3 + 1];
      float x1v2 = x1u[(sn * 64 + u) * 3 + 2];
      float dr = x1v0 * yr[w][m][1] + x1v1 * yr[w][m][2] + x1v2 * yr[w][m][3];
      float di = x1v0 * yi[w][m][1] + x1v1 * yi[w][m][2] + x1v2 * yi[w][m][3];
      atomAdd(&msr[rc * 128 + 64 + u], dr * INV3C * wDv);
      atomAdd(&msi[rc * 128 + 64 + u], di * INV3C * wDv);
      float cr = yr[w][m][0] * wCv;
      float ci = yi[w][m][0] * wCv;
      atomAdd(&mvr[(rc * 128 + 64 + u) * 3 + 0], x1v0 * cr);
      atomAdd(&mvr[(rc * 128 + 64 + u) * 3 + 1], x1v1 * cr);
      atomAdd(&mvr[(rc * 128 + 64 + u) * 3 + 2], x1v2 * cr);
      atomAdd(&mvi[(rc * 128 + 64 + u) * 3 + 0], x1v0 * ci);
      atomAdd(&mvi[(rc * 128 + 64 + u) * 3 + 1], x1v1 * ci);
      atomAdd(&mvi[(rc * 128 + 64 + u) * 3 + 2], x1v2 * ci);
    }
  }
}

// ---------------------------------------------------------------------------
// Kernel 3: output linears (WMMA) + skip add + SiLU gate + interleaved store.
// One wave per 16 nodes, 2 waves / block, no LDS needed.
// ---------------------------------------------------------------------------
__global__ __launch_bounds__(64) void k3_out(
    const float* __restrict__ msr, const float* __restrict__ mvr,
    const float* __restrict__ msi, const float* __restrict__ mvi,
    const float* __restrict__ sc_s, const float* __restrict__ sc_v,
    const float* __restrict__ W_lin_s, const float* __restrict__ W_lin_v,
    float* __restrict__ out, int N)
{
  const int w    = threadIdx.x >> 5;
  const int lane = threadIdx.x & 31;
  const int tile = blockIdx.x * 2 + w;
  const int nT   = (N + 15) >> 4;
  if (tile >= nT) return;

  const int n0    = tile * 16;
  const int col   = lane & 15;
  const int kh    = (lane >> 4) * 2;
  const int mrow  = lane & 15;
  const int rbase = (lane >> 4) * 8;
  const int arow  = (n0 + mrow) * 128;
  const bool full = (n0 + 16 <= N);

  float gR[4][8], gI[4][8];

  // ---- s = ms @ W_lin_s : single K sweep, 8 tiles x {real, imag} ----
  {
    v8f sT[8], iT[8];
#pragma unroll
    for (int t = 0; t < 8; ++t) { sT[t] = zero8(); iT[t] = zero8(); }
    for (int kb = 0; kb < 32; ++kb) {
      int k0 = kb * 4 + kh;
      v2f aR, aI;
      aR.x = msr[arow + k0]; aR.y = msr[arow + k0 + 1];
      aI.x = msi[arow + k0]; aI.y = msi[arow + k0 + 1];
      const float* Bp = W_lin_s + k0 * 128 + col;
#pragma unroll
      for (int t = 0; t < 8; ++t) {
        v2f b; b.x = Bp[t * 16]; b.y = Bp[128 + t * 16];
        sT[t] = wmma4(aR, b, sT[t]);
        iT[t] = wmma4(aI, b, iT[t]);
      }
    }
    // scalar outputs (tiles 0..3) + gate values (tiles 4..7)
#pragma unroll
    for (int j = 0; j < 4; ++j) {
#pragma unroll
      for (int r = 0; r < 8; ++r) {
        int nn = n0 + rbase + r;
        bool ok = full || (nn < N);
        int p = j * 16 + col;
        float skS = ok ? sc_s[nn * 128 + p] : 0.f;
        float skG = ok ? sc_s[nn * 128 + 64 + p] : 0.f;
        float sR = sT[j][r] * LIN_NORM + skS;
        float sI = iT[j][r] * LIN_NORM;
        if (ok) {
          out[nn * 512 + p * 2 + 0] = silu(sR);
          out[nn * 512 + p * 2 + 1] = silu(sI);
        }
        gR[j][r] = silu(sT[4 + j][r] * LIN_NORM + skG);
        gI[j][r] = silu(iT[4 + j][r] * LIN_NORM);
      }
    }
  }

  // ---- v = mv @ W_lin_v per i : K sweep, 4 tiles x {real, imag} ----
  for (int i = 0; i < 3; ++i) {
    v8f vT[4], wT[4];
#pragma unroll
    for (int t = 0; t < 4; ++t) { vT[t] = zero8(); wT[t] = zero8(); }
    for (int kb = 0; kb < 32; ++kb) {
      int k0 = kb * 4 + kh;
      v2f aR, aI;
      aR.x = mvr[(arow + k0) * 3 + i]; aR.y = mvr[(arow + k0 + 1) * 3 + i];
      aI.x = mvi[(arow + k0) * 3 + i]; aI.y = mvi[(arow + k0 + 1) * 3 + i];
      const float* Bp = W_lin_v + k0 * 64 + col;
#pragma unroll
      for (int t = 0; t < 4; ++t) {
        v2f b; b.x = Bp[t * 16]; b.y = Bp[64 + t * 16];
        vT[t] = wmma4(aR, b, vT[t]);
        wT[t] = wmma4(aI, b, wT[t]);
      }
    }
#pragma unroll
    for (int j = 0; j < 4; ++j) {
#pragma unroll
      for (int r = 0; r < 8; ++r) {
        int nn = n0 + rbase + r;
        if (!full && nn >= N) continue;
        int u = j * 16 + col;
        float vR = vT[j][r] * LIN_NORM + sc_v[(nn * 64 + u) * 3 + i];
        float vI = wT[j][r] * LIN_NORM;
        int p = 64 + u * 3 + i;
        out[nn * 512 + p * 2 + 0] = gR[j][r] * vR;
        out[nn * 512 + p * 2 + 1] = gI[j][r] * vI;
      }
    }
  }
}

// ---------------------------------------------------------------------------
extern "C" void kernel_launch(void* const* d_in, const int* in_sizes, int n_in,
                              void* d_out, int out_size, void* d_ws, size_t ws_size,
                              hipStream_t stream) {
  const float* node_attrs = (const float*)d_in[0];
  const float* node_feats = (const float*)d_in[1];
  const float* ear        = (const float*)d_in[2];
  const float* eai        = (const float*)d_in[3];
  const float* edge_feats = (const float*)d_in[4];
  const float* W_up0      = (const float*)d_in[5];
  const float* W_up1      = (const float*)d_in[6];
  const float* M1         = (const float*)d_in[7];
  const float* M2         = (const float*)d_in[8];
  const float* M3         = (const float*)d_in[9];
  const float* M4         = (const float*)d_in[10];
  const float* W_lin_s    = (const float*)d_in[11];
  const float* W_lin_v    = (const float*)d_in[12];
  const float* W_sk_s     = (const float*)d_in[13];
  const float* W_sk_v     = (const float*)d_in[14];
  const int*   eidx       = (const int*)d_in[15];

  const int N = in_sizes[0] / 10;
  const int E = in_sizes[4] / 8;

  // Workspace layout (floats). Total = N*1600 floats (~64 MB for N=10000).
  float* ws = (float*)d_ws;
  size_t off = 0;
  float* x0u  = ws + off; off += (size_t)N * 64;
  float* x1u  = ws + off; off += (size_t)N * 192;
  float* scs  = ws + off; off += (size_t)N * 128;
  float* scv  = ws + off; off += (size_t)N * 192;
  float* msr  = ws + off; off += (size_t)N * 128;
  float* mvr  = ws + off; off += (size_t)N * 384;
  float* msi  = ws + off; off += (size_t)N * 128;
  float* mvi  = ws + off; off += (size_t)N * 384;
  (void)ws_size; (void)n_in; (void)out_size;

  // Zero message accumulators (msr..mvi are contiguous: N*1024 floats).
  (void)hipMemsetAsync(msr, 0, sizeof(float) * (size_t)N * 1024, stream);

  const int nTn = (N + 15) / 16;
  const int nTe = (E + 15) / 16;

  k1_node_prep<<<(nTn + 1) / 2, 64, 0, stream>>>(
      node_attrs, node_feats, W_up0, W_up1, W_sk_s, W_sk_v,
      x0u, x1u, scs, scv, N);

  k2_edge<<<(nTe + 1) / 2, 64, 0, stream>>>(
      edge_feats, ear, eai, eidx, M1, M2, M3, M4,
      x0u, x1u, msr, mvr, msi, mvi, N, E);

  k3_out<<<(nTn + 1) / 2, 64, 0, stream>>>(
      msr, mvr, msi, mvi, scs, scv, W_lin_s, W_lin_v,
      (float*)d_out, N);
}